// LSTM_71416716197888
// MI455X (gfx1250) — compile-verified
//
#include <hip/hip_runtime.h>
#include <hip/hip_bf16.h>

// Fully fused 4-layer LSTM + linear head, persistent kernel, latency-optimized.
// One wave32 owns a 16-element batch tile for all S=2048 timesteps.
// Gate GEMMs: V_WMMA_F32_16X16X4_F32 (A = 16x4 weights resident in VGPRs,
// B = 4x16 batch-striped state, C/D = 16x16 gates accumulating bias).
//
// Latency-shaping tricks:
//  * Gate rows permuted to (i0..3, g0..3 | f0..3, o0..3): sig(i)*tanh(g) is
//    lane-local; lanes 0-15 own cells {0,1}, lanes 16-31 own {2,3} -> new h is
//    produced directly in WMMA B-matrix layout; one v_permlanex16 exchange
//    stage (4 independent swaps) per layer; zero LDS traffic.
//  * Recurrent WMMA pipelined one timestep ahead: P[l] = Whh*h + bias is
//    issued right after h(l,t) is formed and consumed at t+1, so only ONE
//    WMMA sits on the per-layer dependency chain.
//  * Sigmoid pre-scale (0.5) baked into the i/f/o weight rows and biases:
//    activations are v_tanh_f32 directly on the WMMA output + one fma.
//  * x(t+1) load software-pipelined one iteration ahead.

typedef __attribute__((ext_vector_type(2))) float v2f;
typedef __attribute__((ext_vector_type(8))) float v8f;

#define S_LEN 2048
#define BATCH 2048
#define NLAYER 4

__device__ __forceinline__ float xswap16(float v) {
    // exchange lane L <-> lane L+16 (VALU cross-row op; no LDS path)
#if __has_builtin(__builtin_amdgcn_permlanex16)
    int r = __builtin_amdgcn_permlanex16(__float_as_int(v), __float_as_int(v),
                                         0x76543210, 0xfedcba98, false, false);
    return __int_as_float(r);
#else
    return __shfl_xor(v, 16, 32);
#endif
}

__device__ __forceinline__ float fast_tanh(float x) {
#if __has_builtin(__builtin_amdgcn_tanhf)
    return __builtin_amdgcn_tanhf(x);          // v_tanh_f32 (gfx1250 trans op)
#else
    float e = __builtin_amdgcn_exp2f(2.88539008177793f * x);   // e^(2x)
    return 1.0f - 2.0f * __builtin_amdgcn_rcpf(1.0f + e);
#endif
}

// WMMA-output row M -> source weight row (PyTorch gate order i,f,g,o; H=4):
// M 0..3 -> i rows 0..3 ; M 4..7 -> g rows 8..11 ;
// M 8..11 -> f rows 4..7 ; M 12..15 -> o rows 12..15.
__device__ __forceinline__ int gate_perm(int m) {
    return (m < 4) ? m : (m < 8) ? (m + 4) : (m < 12) ? (m - 4) : m;
}

// Rows consumed by a sigmoid (i, f, o) carry a baked-in 0.5 pre-scale;
// tanh rows (g = output rows 4..7) are unscaled.
__device__ __forceinline__ float row_scale(int m) {
    return (m >= 4 && m < 8) ? 1.0f : 0.5f;
}

__global__ __launch_bounds__(32)
void lstm4_fused_wmma(const float* __restrict__ x,
                      const float* __restrict__ Wih0, const float* __restrict__ Whh0,
                      const float* __restrict__ bih0, const float* __restrict__ bhh0,
                      const float* __restrict__ Wihr, const float* __restrict__ Whhr,
                      const float* __restrict__ bihr, const float* __restrict__ bhhr,
                      const float* __restrict__ Wout, const float* __restrict__ bout,
                      float* __restrict__ out)
{
    const int  lane  = threadIdx.x & 31;
    const int  tile  = blockIdx.x;          // 128 tiles of 16 batch elements
    const int  n     = lane & 15;           // batch element within tile
    const bool hi    = lane >= 16;
    const int  bbase = tile * 16;

    // ---------------- Weights in WMMA A-matrix layout (row-permuted) --------
    // A (16x4 f32): lane L holds sc*W[perm(L&15)][k0], k0 = hi?2:0.
    const int   m   = n;
    const int   pr  = gate_perm(m);
    const int   k0  = hi ? 2 : 0;
    const float sc  = row_scale(m);

    v2f A0[NLAYER], A1[NLAYER];
    {
        float w0 = Wih0[pr * 2 + 0];        // layer-0 input weights are [16,2]
        float w1 = Wih0[pr * 2 + 1];        // K=2,3 columns zero-padded
        A0[0].x = hi ? 0.0f : sc * w0;
        A0[0].y = hi ? 0.0f : sc * w1;
        A1[0].x = sc * Whh0[pr * 4 + k0];
        A1[0].y = sc * Whh0[pr * 4 + k0 + 1];
    }
#pragma unroll
    for (int l = 1; l < NLAYER; ++l) {
        const float* wi = Wihr + (size_t)(l - 1) * 16 * 4;
        const float* wh = Whhr + (size_t)(l - 1) * 16 * 4;
        A0[l].x = sc * wi[pr * 4 + k0];  A0[l].y = sc * wi[pr * 4 + k0 + 1];
        A1[l].x = sc * wh[pr * 4 + k0];  A1[l].y = sc * wh[pr * 4 + k0 + 1];
    }

    // Combined bias (b_ih + b_hh), row-permuted + pre-scaled, in C/D layout:
    // VGPR v of lane L corresponds to output row v + (hi?8:0).
    v8f BIASC[NLAYER];
    {
        const int r0 = hi ? 8 : 0;
#pragma unroll
        for (int v = 0; v < 8; ++v) {
            int   src = gate_perm(r0 + v);
            float s   = row_scale(r0 + v);
            BIASC[0][v] = s * (bih0[src] + bhh0[src]);
        }
#pragma unroll
        for (int l = 1; l < NLAYER; ++l) {
#pragma unroll
            for (int v = 0; v < 8; ++v) {
                int   src = gate_perm(r0 + v);
                float s   = row_scale(r0 + v);
                BIASC[l][v] = s * (bihr[(l - 1) * 16 + src] + bhhr[(l - 1) * 16 + src]);
            }
        }
    }

    // Head weights: each half of the wave accumulates its two hidden cells.
    const float wA = hi ? Wout[2] : Wout[0];
    const float wB = hi ? Wout[3] : Wout[1];
    const float bo = bout[0];

    // b_j activation: lanes 0-15 -> tanh(g_j); lanes 16-31 -> sigmoid(o_j).
    // (o rows were pre-scaled by 0.5, so both are one v_tanh + one fma.)
    const float mm = hi ? 0.5f : 1.0f;      // post-scale
    const float cc = hi ? 0.5f : 0.0f;      // post-offset

    // ---------------- Recurrent state (registers only) ----------------
    // hB[l]: layer-l hidden state in WMMA B-matrix layout.
    // cs[l][s]: cell state; lane owns cells { (hi?2:0)+s : s=0,1 }.
    v2f   hB[NLAYER];
    float cs[NLAYER][2];
#pragma unroll
    for (int l = 0; l < NLAYER; ++l) {
        hB[l].x = 0.0f; hB[l].y = 0.0f;
        cs[l][0] = 0.0f; cs[l][1] = 0.0f;
    }

    // Pipelined recurrent partial: P[l] = Whh*h(l, t-1) + bias, ready for t.
    v8f P[NLAYER];
#pragma unroll
    for (int l = 0; l < NLAYER; ++l)
        P[l] = __builtin_amdgcn_wmma_f32_16x16x4_f32(
                   false, A1[l], false, hB[l], (short)0, BIASC[l], false, false);

    const float* xq = x + ((size_t)bbase + n) * 2;     // x[t][bbase+n][:]
    float*       op = out + bbase + n;                 // y[t][bbase+n]

    float cx0 = xq[0], cx1 = xq[1];                    // x(t=0), preloaded

    for (int t = 0; t < S_LEN; ++t) {
        // -------- software-pipelined x prefetch: issue load for t+1 now ----
        const float* xn = (t + 1 < S_LEN) ? (xq + (size_t)BATCH * 2) : xq;
        float nx0 = xn[0], nx1 = xn[1];

        // Layer-0 input in B-matrix layout: K=0,1 = (x0,x1), K=2,3 = 0.
        v2f zcur;
        zcur.x = hi ? 0.0f : cx0;
        zcur.y = hi ? 0.0f : cx1;

        float hA0 = 0.f, hB0 = 0.f;   // last layer's owned hidden cells

#pragma unroll
        for (int l = 0; l < NLAYER; ++l) {
            // gates = Wih*z + P[l]   (single WMMA on the dependency chain;
            // P[l] = Whh*h + bias was issued during the previous timestep)
            v8f d = __builtin_amdgcn_wmma_f32_16x16x4_f32(
                        false, A0[l], false, zcur, (short)0, P[l], false, false);

            // rows seen: lanes<16 -> i0..3, g0..3 ; lanes>=16 -> f0..3, o0..3
            float a0 = fmaf(0.5f, fast_tanh(d[0]), 0.5f);  // lo: sig(i), hi: sig(f)
            float a1 = fmaf(0.5f, fast_tanh(d[1]), 0.5f);
            float a2 = fmaf(0.5f, fast_tanh(d[2]), 0.5f);
            float a3 = fmaf(0.5f, fast_tanh(d[3]), 0.5f);
            float b0 = fmaf(mm, fast_tanh(d[4]), cc);      // lo: tanh(g), hi: sig(o)
            float b1 = fmaf(mm, fast_tanh(d[5]), cc);
            float b2 = fmaf(mm, fast_tanh(d[6]), cc);
            float b3 = fmaf(mm, fast_tanh(d[7]), cc);

            // lane-local input-gate products (meaningful in lanes 0-15)
            float ig0 = a0 * b0, ig1 = a1 * b1, ig2 = a2 * b2, ig3 = a3 * b3;

            // single cross-half exchange stage (4 independent swaps):
            //   lo sends ig2,ig3 / receives sig(f0),sig(f1),sig(o0),sig(o1)
            //   hi sends sig(f0..1),sig(o0..1) / receives ig2,ig3
            float u0 = hi ? a0 : ig2;
            float u1 = hi ? a1 : ig3;
            float w0 = hi ? b0 : 0.0f;
            float w1 = hi ? b1 : 0.0f;
            float su0 = xswap16(u0);
            float su1 = xswap16(u1);
            float sw0 = xswap16(w0);
            float sw1 = xswap16(w1);

            // per-lane owned cells: slot s -> hidden index (hi?2:0)+s
            float sf0 = hi ? a2 : su0;      // sig(f) for owned cell 0
            float sf1 = hi ? a3 : su1;
            float so0 = hi ? b2 : sw0;      // sig(o) for owned cell 0
            float so1 = hi ? b3 : sw1;
            float igA = hi ? su0 : ig0;     // sig(i)*tanh(g) for owned cell 0
            float igB = hi ? su1 : ig1;

            float c0n = fmaf(sf0, cs[l][0], igA);
            float c1n = fmaf(sf1, cs[l][1], igB);
            cs[l][0] = c0n;
            cs[l][1] = c1n;
            float h0n = so0 * fast_tanh(c0n);
            float h1n = so1 * fast_tanh(c1n);

            // already in B-matrix layout: lo lanes carry K0,K1 = h0,h1;
            // hi lanes carry K2,K3 = h2,h3. No repack shuffle needed.
            v2f nb; nb.x = h0n; nb.y = h1n;
            hB[l] = nb;
            zcur  = nb;                     // next layer's input

            // issue next timestep's recurrent partial NOW (off critical path)
            P[l] = __builtin_amdgcn_wmma_f32_16x16x4_f32(
                       false, A1[l], false, nb, (short)0, BIASC[l], false, false);

            if (l == NLAYER - 1) { hA0 = h0n; hB0 = h1n; }
        }

        // Linear head: each half reduces its two cells, one swap combines.
        float part = fmaf(wA, hA0, wB * hB0);
        float y = part + xswap16(part) + bo;
        if (!hi) *op = y;
        op += BATCH;

        cx0 = nx0; cx1 = nx1; xq = xn;      // rotate pipelined x
    }
}

extern "C" void kernel_launch(void* const* d_in, const int* in_sizes, int n_in,
                              void* d_out, int out_size, void* d_ws, size_t ws_size,
                              hipStream_t stream) {
    (void)in_sizes; (void)n_in; (void)out_size; (void)d_ws; (void)ws_size;
    const float* x    = (const float*)d_in[0];
    const float* Wih0 = (const float*)d_in[1];
    const float* Whh0 = (const float*)d_in[2];
    const float* bih0 = (const float*)d_in[3];
    const float* bhh0 = (const float*)d_in[4];
    const float* Wihr = (const float*)d_in[5];
    const float* Whhr = (const float*)d_in[6];
    const float* bihr = (const float*)d_in[7];
    const float* bhhr = (const float*)d_in[8];
    const float* Wout = (const float*)d_in[9];
    const float* bout = (const float*)d_in[10];
    float* out = (float*)d_out;

    dim3 grid(BATCH / 16);   // 128 single-wave workgroups -> spread across WGPs
    dim3 block(32);
    lstm4_fused_wmma<<<grid, block, 0, stream>>>(
        x, Wih0, Whh0, bih0, bhh0, Wihr, Whhr, bihr, bhhr, Wout, bout, out);
}